// YOLOv3Plus_6743098654960
// MI455X (gfx1250) — compile-verified
//
#include <hip/hip_runtime.h>
#include <hip/hip_bf16.h>
#include <math.h>

// ---------------------------------------------------------------------------
// YOLOv3+ decode + class-aware greedy NMS for MI455X (gfx1250, wave32).
//
// Roofline: 8.6MB in / 0.7MB out @ 23.3TB/s -> memory is free; cost is the
// serial NMS. Plan: parallel decode (WMMA row-sum for softmax denominator),
// 256KB-LDS bitonic argsort (CDNA5 320KB LDS), serial-outer/parallel-inner
// NMS with LDS suppression bitmask (ds atomics) on one workgroup.
// ---------------------------------------------------------------------------

typedef __attribute__((ext_vector_type(2))) float v2f;
typedef __attribute__((ext_vector_type(8))) float v8f;

#define NBOX   25200
#define NPAD   32768          // next pow2 >= NBOX for bitonic sort
#define NCLS   80
#define CONF_T 0.001f
#define NMS_T  0.6f
#define INV_IMG (1.0f/640.0f)

__device__ __constant__ float d_anchors[18] = {
    10.f,13.f, 16.f,30.f, 33.f,23.f,      // stride 8
    30.f,61.f, 62.f,45.f, 59.f,119.f,     // stride 16
    116.f,90.f, 156.f,198.f, 373.f,326.f  // stride 32
};

__device__ __forceinline__ float sigm(float x) { return 1.0f / (1.0f + __expf(-x)); }

// ---------------------------------------------------------------------------
// Kernel 1: decode. One wave32 handles 16 boxes.
//   lane pair (m, m+16) owns box m: lane<16 holds classes c%4 in {0,1},
//   lane>=16 holds c%4 in {2,3}  ==  exactly the V_WMMA f32 16x16x4 A layout.
//   Softmax denominator = row-sum of exp computed as D = A(16x4) * Ones(4x16),
//   accumulated over 20 K-chunks on the matrix pipe.
// ---------------------------------------------------------------------------
__global__ void decode_kernel(const float* __restrict__ ps,
                              const float* __restrict__ pm,
                              const float* __restrict__ pl,
                              float* __restrict__ out)
{
    const int lane = threadIdx.x & 31;
    const int wave = threadIdx.x >> 5;
    const int m    = lane & 15;
    const int box_base = (blockIdx.x * (blockDim.x >> 5) + wave) * 16;

    int g = box_base + m;
    if (g >= NBOX) g = NBOX - 1;      // clamp loads; WMMA needs EXEC all-ones

    // level select (S: 19200 boxes, M: 4800, L: 1200)
    const float* base; int HW, W, lvl, local; float stp;
    if (g < 19200)      { base = ps; HW = 6400; W = 80; stp = 8.f;  lvl = 0; local = g; }
    else if (g < 24000) { base = pm; HW = 1600; W = 40; stp = 16.f; lvl = 1; local = g - 19200; }
    else                { base = pl; HW = 400;  W = 20; stp = 32.f; lvl = 2; local = g - 24000; }
    const int cell = local / 3;
    const int a    = local - cell * 3;
    const int gx   = cell % W;
    const int gy   = cell / W;

    // ---- load this lane's 40 class logits (classes 4k+off, 4k+off+1) ----
    const float* cbase = base + (size_t)(3 + a * NCLS) * HW + cell;
    const int off = (lane < 16) ? 0 : 2;
    float cv[40];
#pragma unroll
    for (int k = 0; k < 20; ++k) {
        cv[2 * k]     = cbase[(size_t)(4 * k + off)     * HW];
        cv[2 * k + 1] = cbase[(size_t)(4 * k + off + 1) * HW];
    }

    // ---- max / argmax (first-occurrence semantics like jnp.argmax) ----
    float mx = cv[0]; int mi = off;
#pragma unroll
    for (int k = 0; k < 20; ++k) {
#pragma unroll
        for (int j = 0; j < 2; ++j) {
            float v = cv[2 * k + j]; int c = 4 * k + off + j;
            if (v > mx) { mx = v; mi = c; }
        }
    }
    {   // merge the two half-lanes of the pair (wave32 xor-shuffle)
        float omx = __shfl_xor(mx, 16, 32);
        int   omi = __shfl_xor(mi, 16, 32);
        if (omx > mx || (omx == mx && omi < mi)) { mx = omx; mi = omi; }
    }

    // ---- softmax denominator via WMMA row-sum: acc += exp(A) * Ones ----
    float esum;
    v8f acc = {0.f,0.f,0.f,0.f,0.f,0.f,0.f,0.f};
#if __has_builtin(__builtin_amdgcn_wmma_f32_16x16x4_f32)
    v2f B; B.x = 1.0f; B.y = 1.0f;     // all-ones B => D columns = row sums
#pragma unroll
    for (int k = 0; k < 20; ++k) {
        v2f A;
        A.x = __expf(cv[2 * k]     - mx);
        A.y = __expf(cv[2 * k + 1] - mx);
        acc = __builtin_amdgcn_wmma_f32_16x16x4_f32(
                  false, A, false, B, (short)0, acc, false, false);
    }
    // D layout: lanes 0-15 VGPR r -> row r; lanes 16-31 VGPR r -> row 8+r.
    // Box m's sum lives at lane (m<8 ? m : m+16), register (m&7).
    {
        int r = lane & 7;
        float sel = acc[0];
        if (r == 1) sel = acc[1];
        if (r == 2) sel = acc[2];
        if (r == 3) sel = acc[3];
        if (r == 4) sel = acc[4];
        if (r == 5) sel = acc[5];
        if (r == 6) sel = acc[6];
        if (r == 7) sel = acc[7];
        int src = (m < 8) ? m : (m + 16);
        esum = __shfl(sel, src, 32);
    }
#else
    {   // fallback: VALU pair-sum (kept only so compile never fails)
        float part = 0.f;
#pragma unroll
        for (int k = 0; k < 40; ++k) part += __expf(cv[k] - mx);
        esum = part + __shfl_xor(part, 16, 32);
    }
#endif

    // ---- objectness + box regression ----
    const float obj = base[(size_t)a * HW + cell];
    const float* rbase = base + (size_t)(3 + 3 * NCLS + a * 4) * HW + cell; // ch 243+
    const float r0 = rbase[0];
    const float r1 = rbase[(size_t)HW];
    const float r2 = rbase[(size_t)2 * HW];
    const float r3 = rbase[(size_t)3 * HW];

    // max(softmax)*sigmoid(obj) == sigmoid(obj) / sum(exp(x - xmax))
    const float score = sigm(obj) / esum;

    const float cx = (sigm(r0) + (float)gx) * stp;
    const float cy = (sigm(r1) + (float)gy) * stp;
    const float bw = __expf(r2) * d_anchors[lvl * 6 + a * 2 + 0];
    const float bh = __expf(r3) * d_anchors[lvl * 6 + a * 2 + 1];
    const float x1 = fminf(fmaxf((cx - 0.5f * bw) * INV_IMG, 0.f), 1.f);
    const float y1 = fminf(fmaxf((cy - 0.5f * bh) * INV_IMG, 0.f), 1.f);
    const float x2 = fminf(fmaxf((cx + 0.5f * bw) * INV_IMG, 0.f), 1.f);
    const float y2 = fminf(fmaxf((cy + 0.5f * bh) * INV_IMG, 0.f), 1.f);

    if (lane < 16 && (box_base + m) < NBOX) {
        float* o = out + (size_t)(box_base + m) * 7;
        o[0] = x1; o[1] = y1; o[2] = x2; o[3] = y2;
        o[4] = score; o[5] = (float)mi;           // o[6] (keep) set by NMS
    }
}

// ---------------------------------------------------------------------------
// Kernel 2: stable argsort(-score) via bitonic sort of packed u64 keys,
// entirely inside 256KB of CDNA5 LDS (one WGP). key = (~score_bits<<32)|idx:
// ascending u64 sort == descending score, ascending index tie-break.
// ---------------------------------------------------------------------------
__global__ void sort_kernel(const float* __restrict__ out, int* __restrict__ sidx)
{
    extern __shared__ unsigned long long sm[];
    const int tid = threadIdx.x, nth = blockDim.x;

    for (int i = tid; i < NPAD; i += nth) {
        unsigned ks = 0u;                                // pads sort last
        if (i < NBOX) ks = __float_as_uint(out[(size_t)i * 7 + 4]); // score>0
        sm[i] = ((unsigned long long)(~ks) << 32) | (unsigned)i;
    }
    for (unsigned k = 2; k <= NPAD; k <<= 1) {
        for (unsigned j = k >> 1; j > 0; j >>= 1) {
            __syncthreads();
            for (unsigned i = tid; i < NPAD; i += nth) {
                unsigned p = i ^ j;
                if (p > i) {
                    unsigned long long va = sm[i], vb = sm[p];
                    bool up = ((i & k) == 0);
                    if (up ? (va > vb) : (va < vb)) { sm[i] = vb; sm[p] = va; }
                }
            }
        }
    }
    __syncthreads();
    for (int i = tid; i < NBOX; i += nth)
        sidx[i] = (int)(sm[i] & 0xFFFFFFFFull);
}

// ---------------------------------------------------------------------------
// Kernel 3: gather boxes (with class offset, as in reference) + scores into
// sorted order for coalesced NMS inner loops.
// ---------------------------------------------------------------------------
__global__ void gather_kernel(const float* __restrict__ out,
                              const int* __restrict__ sidx,
                              float4* __restrict__ sbox,
                              float* __restrict__ sscore)
{
    int i = blockIdx.x * blockDim.x + threadIdx.x;
    if (i >= NBOX) return;
    const float* row = out + (size_t)sidx[i] * 7;
    float offc = row[5] * 2.0f;                 // class-aware offset trick
    sbox[i]   = make_float4(row[0] + offc, row[1] + offc,
                            row[2] + offc, row[3] + offc);
    sscore[i] = row[4];
}

// ---------------------------------------------------------------------------
// Kernel 4: greedy NMS. Serial outer loop (true data dependence), parallel
// inner IoU sweep. Suppression bitmask in LDS via ds_or atomics. A box is
// suppressed iff it overlaps a previously KEPT box == reference semantics.
// ---------------------------------------------------------------------------
__global__ void nms_kernel(const float4* __restrict__ sbox,
                           const float* __restrict__ sscore,
                           const int* __restrict__ sidx,
                           float* __restrict__ out)
{
    __shared__ unsigned supp[(NBOX + 31) / 32];
    const int tid = threadIdx.x, nth = blockDim.x;
    for (int w = tid; w < (NBOX + 31) / 32; w += nth) supp[w] = 0u;
    __syncthreads();

    for (int i = 0; i < NBOX; ++i) {
        // bit i only ever written in earlier (barriered) iterations -> stable
        bool sup_i  = (supp[i >> 5] >> (i & 31)) & 1u;
        bool active = (!sup_i) && (sscore[i] >= CONF_T);   // uniform
        if (tid == 0) out[(size_t)sidx[i] * 7 + 6] = active ? 1.0f : 0.0f;
        if (active) {
            const float4 bi = sbox[i];
            const float  ai = (bi.z - bi.x) * (bi.w - bi.y);
            for (int j = i + 1 + tid; j < NBOX; j += nth) {
                if (j + nth < NBOX)
                    __builtin_prefetch((const void*)(sbox + j + nth), 0, 0);
                float4 bj = sbox[j];
                float xx1 = fmaxf(bi.x, bj.x), yy1 = fmaxf(bi.y, bj.y);
                float xx2 = fminf(bi.z, bj.z), yy2 = fminf(bi.w, bj.w);
                float w = fmaxf(1e-28f, xx2 - xx1);
                float h = fmaxf(1e-28f, yy2 - yy1);
                float inter = w * h;
                float aj = (bj.z - bj.x) * (bj.w - bj.y);
                // iou > t  <=>  inter > t*(ai+aj-inter)   (division-free)
                if (inter > NMS_T * (ai + aj - inter))
                    atomicOr(&supp[j >> 5], 1u << (j & 31));
            }
            __syncthreads();   // uniform: all threads agree on `active`
        }
    }
}

// ---------------------------------------------------------------------------
extern "C" void kernel_launch(void* const* d_in, const int* in_sizes, int n_in,
                              void* d_out, int out_size, void* d_ws, size_t ws_size,
                              hipStream_t stream)
{
    const float* ps = (const float*)d_in[0];   // (1,255,80,80)
    const float* pm = (const float*)d_in[1];   // (1,255,40,40)
    const float* pl = (const float*)d_in[2];   // (1,255,20,20)
    float* out = (float*)d_out;                // (25200, 7)

    // workspace layout (~635 KB)
    char* ws = (char*)d_ws;
    int*    sidx   = (int*)ws;                          // 32768*4  = 131072 B
    float4* sbox   = (float4*)(ws + 131072);            // 25200*16 = 403200 B
    float*  sscore = (float*)(ws + 131072 + 403200);    // 25200*4  = 100800 B

    // 1) decode: 16 boxes/wave, 8 waves/block -> 128 boxes/block
    decode_kernel<<<(NBOX + 127) / 128, 256, 0, stream>>>(ps, pm, pl, out);
    // 2) stable sort by score desc, 256KB dynamic LDS (CDNA5 320KB/WGP)
    sort_kernel<<<1, 1024, NPAD * sizeof(unsigned long long), stream>>>(out, sidx);
    // 3) gather sorted box data
    gather_kernel<<<(NBOX + 255) / 256, 256, 0, stream>>>(out, sidx, sbox, sscore);
    // 4) greedy NMS (serial-outer / parallel-inner), writes keep column
    nms_kernel<<<1, 1024, 0, stream>>>(sbox, sscore, sidx, out);
}